// EGNN_48507360641326
// MI455X (gfx1250) — compile-verified
//
#include <hip/hip_runtime.h>
#include <math.h>

typedef float v2f __attribute__((ext_vector_type(2)));
typedef float v4f __attribute__((ext_vector_type(4)));
typedef float v8f __attribute__((ext_vector_type(8)));

#define N_NODES 50000
#define N_EDGES 800000
#define FDIM 64
#define LSTRIDE 68   // 16x64 LDS tile padded: stride 68 floats -> conflict-free column reads

__device__ __forceinline__ float silu_f(float v) {
    return v * (1.0f / (1.0f + __expf(-v)));
}

__device__ __forceinline__ v8f wmma4(v2f a, v2f b, v8f c) {
    // V_WMMA_F32_16X16X4_F32 : D = A(16x4) * B(4x16) + C(16x16), fp32 all the way
    return __builtin_amdgcn_wmma_f32_16x16x4_f32(false, a, false, b, (short)0, c, false, false);
}

// One K-step against packed weights: two b128 loads supply the four 16-col
// B fragments; A fragment `a` is shared. acc[0..3] are the 4 column tiles.
__device__ __forceinline__ void wmma_step(const float* __restrict__ Wp, int t, int lane,
                                          v2f a, v8f* acc) {
    const float* p = Wp + ((size_t)(t * 32 + lane) << 3);
    const v4f f0 = *(const v4f*)(p);
    const v4f f1 = *(const v4f*)(p + 4);
    v2f b;
    b.x = f0.x; b.y = f0.y; acc[0] = wmma4(a, b, acc[0]);
    b.x = f0.z; b.y = f0.w; acc[1] = wmma4(a, b, acc[1]);
    b.x = f1.x; b.y = f1.y; acc[2] = wmma4(a, b, acc[2]);
    b.x = f1.z; b.y = f1.w; acc[3] = wmma4(a, b, acc[3]);
}

// acc += tile(16x64 in LDS, row stride LSTRIDE) @ Wpacked(64x64)
__device__ __forceinline__ void gemm_lds(const float* __restrict__ tile,
                                         const float* __restrict__ Wp,
                                         int m, int half, int lane, v8f* acc) {
    #pragma unroll
    for (int t = 0; t < 16; ++t) {
        const int k0 = 4 * t + 2 * half;
        v2f a;
        a.x = tile[m * LSTRIDE + k0];
        a.y = tile[m * LSTRIDE + k0 + 1];
        wmma_step(Wp, t, lane, a, acc);
    }
}

// Reorder W[krows x 64] (row-major) into WMMA B-fragment order:
// dst[(t*32+lane)*8 + q],  q = j*2+v  ->  W[4t + 2*(lane/16) + v][j*16 + lane%16]
// rows >= krows packed as zero (K padding).
__global__ void pack_kernel(const float* __restrict__ W, float* __restrict__ dst,
                            int krows, int T) {
    const int idx = blockIdx.x * blockDim.x + threadIdx.x;
    if (idx >= T * 256) return;
    const int t    = idx >> 8;
    const int r    = idx & 255;
    const int lane = r >> 3;
    const int q    = r & 7;
    const int j    = q >> 1;
    const int v    = q & 1;
    const int k    = 4 * t + 2 * (lane >> 4) + v;
    dst[idx] = (k < krows) ? W[k * FDIM + j * 16 + (lane & 15)] : 0.0f;
}

// One wave32 per tile of 16 edges. Fused: edge MLP -> pos MLP -> scatter atomics.
__global__ void __launch_bounds__(32)
egnn_edge_kernel(const float* __restrict__ x, const float* __restrict__ h,
                 const int* __restrict__ edges,
                 const float* __restrict__ We1p, const float* __restrict__ be1,
                 const float* __restrict__ We2p, const float* __restrict__ be2,
                 const float* __restrict__ Wp1p, const float* __restrict__ bp1,
                 const float* __restrict__ Wp2, const float* __restrict__ bp2,
                 float* __restrict__ cnt, float* __restrict__ xsum,
                 float* __restrict__ msum)
{
    __shared__ float tile[16 * LSTRIDE];
    const int lane = threadIdx.x & 31;
    const int m    = lane & 15;      // A row / B column lane index
    const int half = lane >> 4;
    const int n    = m;
    const int e0   = blockIdx.x * 16;
    const int e    = e0 + m;

    const int src = edges[e];
    const int dst = edges[N_EDGES + e];
    const float dx = x[src * 3 + 0] - x[dst * 3 + 0];
    const float dy = x[src * 3 + 1] - x[dst * 3 + 1];
    const float dz = x[src * 3 + 2] - x[dst * 3 + 2];
    const float dij = sqrtf(dx * dx + dy * dy + dz * dz);

    const float* __restrict__ hs = h + (size_t)src * FDIM;
    const float* __restrict__ hd = h + (size_t)dst * FDIM;

    // ---- edge layer 1: [16 x 129(pad 132)] @ We1[129 x 64] ----
    v8f acc[4] = {};
    #pragma unroll 4
    for (int t = 0; t < 33; ++t) {
        const int k0 = 4 * t + 2 * half;
        v2f a;
        if (t < 16)      { a.x = hs[k0];      a.y = hs[k0 + 1]; }
        else if (t < 32) { a.x = hd[k0 - 64]; a.y = hd[k0 - 63]; }
        else             { a.x = (half == 0) ? dij : 0.0f; a.y = 0.0f; }
        wmma_step(We1p, t, lane, a, acc);
    }
    // D element (m,n): lane = (m/8)*16 + n, vgpr = m%8
    #pragma unroll
    for (int j = 0; j < 4; ++j)
        #pragma unroll
        for (int v = 0; v < 8; ++v)
            tile[(v + half * 8) * LSTRIDE + j * 16 + n] = silu_f(acc[j][v] + be1[j * 16 + n]);
    __syncthreads();

    // ---- edge layer 2: @ We2 -> m_ij ----
    v8f macc[4] = {};
    gemm_lds(tile, We2p, m, half, lane, macc);
    #pragma unroll
    for (int j = 0; j < 4; ++j)
        #pragma unroll
        for (int v = 0; v < 8; ++v)
            macc[j][v] = silu_f(macc[j][v] + be2[j * 16 + n]);
    __syncthreads();
    #pragma unroll
    for (int j = 0; j < 4; ++j)
        #pragma unroll
        for (int v = 0; v < 8; ++v)
            tile[(v + half * 8) * LSTRIDE + j * 16 + n] = macc[j][v];
    __syncthreads();

    // ---- pos layer 1: m_ij @ Wp1 ----
    v8f pacc[4] = {};
    gemm_lds(tile, Wp1p, m, half, lane, pacc);
    __syncthreads();
    #pragma unroll
    for (int j = 0; j < 4; ++j)
        #pragma unroll
        for (int v = 0; v < 8; ++v)
            tile[(v + half * 8) * LSTRIDE + j * 16 + n] = silu_f(pacc[j][v] + bp1[j * 16 + n]);
    __syncthreads();

    // ---- pos layer 2: [16x64] @ Wp2[64x1] (per-lane dot, cross-half combine) ----
    float partial = 0.0f;
    #pragma unroll 8
    for (int k = 0; k < 32; ++k) {
        const int kk = half * 32 + k;
        partial += tile[m * LSTRIDE + kk] * Wp2[kk];
    }
    const float w = partial + __shfl_xor(partial, 16, 32) + bp2[0];

    // ---- scatter: x_ij = (x[dst]-x[src])*w = -d*w ; counts ; m_sum ----
    if (half == 0) {
        unsafeAtomicAdd(&xsum[src * 3 + 0], -dx * w);
        unsafeAtomicAdd(&xsum[src * 3 + 1], -dy * w);
        unsafeAtomicAdd(&xsum[src * 3 + 2], -dz * w);
        unsafeAtomicAdd(&cnt[src], 1.0f);
    }
    #pragma unroll
    for (int v = 0; v < 8; ++v) {
        const int mr = v + half * 8;
        const int srcRow = __shfl(src, mr, 32);   // src index of row mr lives in lane mr
        #pragma unroll
        for (int j = 0; j < 4; ++j)
            unsafeAtomicAdd(&msum[(size_t)srcRow * FDIM + j * 16 + n], macc[j][v]);
    }
}

// One wave32 per tile of 16 nodes: x' and node MLP.
__global__ void __launch_bounds__(32)
egnn_node_kernel(const float* __restrict__ x, const float* __restrict__ h,
                 const float* __restrict__ Wn1p, const float* __restrict__ bn1,
                 const float* __restrict__ Wn2p, const float* __restrict__ bn2,
                 const float* __restrict__ cnt, const float* __restrict__ xsum,
                 const float* __restrict__ msum,
                 float* __restrict__ xout, float* __restrict__ hout)
{
    __shared__ float tile[16 * LSTRIDE];
    const int lane = threadIdx.x & 31;
    const int m    = lane & 15;
    const int half = lane >> 4;
    const int n    = m;
    const int i0   = blockIdx.x * 16;
    const int i    = i0 + m;

    const float inv = 1.0f / fmaxf(cnt[i], 1.0f);

    if (half == 0) {
        #pragma unroll
        for (int d = 0; d < 3; ++d)
            xout[i * 3 + d] = x[i * 3 + d] + xsum[i * 3 + d] * inv;
    }

    const float* __restrict__ hrow = h + (size_t)i * FDIM;
    const float* __restrict__ mrow = msum + (size_t)i * FDIM;

    // ---- node layer 1: [16 x 128] @ Wn1[128 x 64]   (cols 0..63 = h, 64..127 = m_i) ----
    v8f acc[4] = {};
    #pragma unroll 4
    for (int t = 0; t < 32; ++t) {
        const int k0 = 4 * t + 2 * half;
        v2f a;
        if (t < 16) { a.x = hrow[k0];            a.y = hrow[k0 + 1]; }
        else        { a.x = mrow[k0 - 64] * inv; a.y = mrow[k0 - 63] * inv; }
        wmma_step(Wn1p, t, lane, a, acc);
    }
    #pragma unroll
    for (int j = 0; j < 4; ++j)
        #pragma unroll
        for (int v = 0; v < 8; ++v)
            tile[(v + half * 8) * LSTRIDE + j * 16 + n] = silu_f(acc[j][v] + bn1[j * 16 + n]);
    __syncthreads();

    // ---- node layer 2: @ Wn2 ; h' = h + out + bn2 ----
    v8f acc2[4] = {};
    gemm_lds(tile, Wn2p, m, half, lane, acc2);
    #pragma unroll
    for (int v = 0; v < 8; ++v) {
        const int mr = v + half * 8;
        const size_t base = (size_t)(i0 + mr) * FDIM;
        #pragma unroll
        for (int j = 0; j < 4; ++j) {
            const int col = j * 16 + n;
            hout[base + col] = h[base + col] + acc2[j][v] + bn2[col];
        }
    }
}

__global__ void zero_kernel(float* __restrict__ p, int count) {
    const int idx = blockIdx.x * blockDim.x + threadIdx.x;
    if (idx < count) p[idx] = 0.0f;
}

extern "C" void kernel_launch(void* const* d_in, const int* in_sizes, int n_in,
                              void* d_out, int out_size, void* d_ws, size_t ws_size,
                              hipStream_t stream) {
    const float* x     = (const float*)d_in[0];
    const float* h     = (const float*)d_in[1];
    const int*   edges = (const int*)d_in[2];
    const float* We1   = (const float*)d_in[3];
    const float* be1   = (const float*)d_in[4];
    const float* We2   = (const float*)d_in[5];
    const float* be2   = (const float*)d_in[6];
    const float* Wp1   = (const float*)d_in[7];
    const float* bp1   = (const float*)d_in[8];
    const float* Wp2   = (const float*)d_in[9];
    const float* bp2   = (const float*)d_in[10];
    const float* Wn1   = (const float*)d_in[11];
    const float* bn1   = (const float*)d_in[12];
    const float* Wn2   = (const float*)d_in[13];
    const float* bn2   = (const float*)d_in[14];

    float* cnt  = (float*)d_ws;          // [N]
    float* xsum = cnt + N_NODES;         // [N,3]
    float* msum = xsum + 3 * N_NODES;    // [N,64]
    float* wpk  = msum + (size_t)N_NODES * FDIM;   // packed weights
    float* We1p = wpk;                   // 33*256
    float* We2p = We1p + 33 * 256;       // 16*256
    float* Wp1p = We2p + 16 * 256;       // 16*256
    float* Wn1p = Wp1p + 16 * 256;       // 32*256
    float* Wn2p = Wn1p + 32 * 256;       // 16*256

    float* xout = (float*)d_out;         // [N,3]
    float* hout = xout + 3 * N_NODES;    // [N,64]

    const int zcount = N_NODES * (1 + 3 + FDIM);
    zero_kernel<<<(zcount + 255) / 256, 256, 0, stream>>>(cnt, zcount);

    pack_kernel<<<(33 * 256 + 255) / 256, 256, 0, stream>>>(We1, We1p, 129, 33);
    pack_kernel<<<(16 * 256 + 255) / 256, 256, 0, stream>>>(We2, We2p, 64, 16);
    pack_kernel<<<(16 * 256 + 255) / 256, 256, 0, stream>>>(Wp1, Wp1p, 64, 16);
    pack_kernel<<<(32 * 256 + 255) / 256, 256, 0, stream>>>(Wn1, Wn1p, 128, 32);
    pack_kernel<<<(16 * 256 + 255) / 256, 256, 0, stream>>>(Wn2, Wn2p, 64, 16);

    egnn_edge_kernel<<<N_EDGES / 16, 32, 0, stream>>>(
        x, h, edges, We1p, be1, We2p, be2, Wp1p, bp1, Wp2, bp2, cnt, xsum, msum);

    egnn_node_kernel<<<N_NODES / 16, 32, 0, stream>>>(
        x, h, Wn1p, bn1, Wn2p, bn2, cnt, xsum, msum, xout, hout);
}